// SelectTrainSphereConv_70025146794018
// MI455X (gfx1250) — compile-verified
//
#include <hip/hip_runtime.h>
#include <hip/hip_bf16.h>

#define B_   4
#define CIN  64
#define COUT 64
#define V_   163842
#define K_   7
#define RR   256            // B*CIN
#define KTOT 448            // CIN*K
#define KSTEPS 112          // KTOT/4
#define ITP_STRIDE 452      // padded row stride (floats) -> conflict-free b64 frag loads
#define WAVES 4
#define NGROUPS ((V_ + 3) / 4)

typedef float v2f __attribute__((ext_vector_type(2)));
typedef float v8f __attribute__((ext_vector_type(8)));

#define WFRAG_BYTES (KSTEPS * 4 * 32 * 8)        // 114688: W in B-fragment order
#define ITPS_FLOATS (16 * ITP_STRIDE)            // 7232 floats per wave
#define ITPS_BYTES  (WAVES * ITPS_FLOATS * 4)    // 115712
#define ITPM_BYTES  (WAVES * 4 * 49 * 4)         // 3136
#define IDX_BYTES   (WAVES * 28 * 4)             // 448
#define SMEM_BYTES  (WFRAG_BYTES + ITPS_BYTES + ITPM_BYTES + IDX_BYTES)  // 233984

// Wave-private LDS ordering: CDNA5 keeps a wave's DS ops in order; we only need
// to stop the compiler from reordering across the phase boundary.
__device__ __forceinline__ void wave_sync() {
  __builtin_amdgcn_wave_barrier();
  asm volatile("" ::: "memory");
}

// ---------------- transpose x (B,C,V) -> xt (V, 256) ----------------
__global__ __launch_bounds__(256) void transpose_in_kernel(
    const float* __restrict__ x, float* __restrict__ xt)
{
  __shared__ float tile[32][33];
  const int v0 = blockIdx.x * 32, r0 = blockIdx.y * 32;
  const int tx = threadIdx.x, ty = threadIdx.y;
#pragma unroll
  for (int j = 0; j < 32; j += 8) {
    int v = v0 + tx, r = r0 + ty + j;
    tile[ty + j][tx] = (v < V_) ? x[(size_t)r * V_ + v] : 0.f;
  }
  __syncthreads();
#pragma unroll
  for (int j = 0; j < 32; j += 8) {
    int v = v0 + ty + j, r = r0 + tx;
    if (v < V_) xt[(size_t)v * RR + r] = tile[tx][ty + j];
  }
}

// ------------- transpose outws (V,256) -> out (B,64,V) + bias -------------
__global__ __launch_bounds__(256) void transpose_out_kernel(
    const float* __restrict__ outws, const float* __restrict__ bias,
    float* __restrict__ out)
{
  __shared__ float tile[32][33];
  const int v0 = blockIdx.x * 32, r0 = blockIdx.y * 32;
  const int tx = threadIdx.x, ty = threadIdx.y;
#pragma unroll
  for (int j = 0; j < 32; j += 8) {
    int v = v0 + ty + j, r = r0 + tx;
    tile[ty + j][tx] = (v < V_) ? outws[(size_t)v * RR + r] : 0.f;
  }
  __syncthreads();
#pragma unroll
  for (int j = 0; j < 32; j += 8) {
    int r = r0 + ty + j, v = v0 + tx;
    if (v < V_) out[(size_t)r * V_ + v] = tile[tx][ty + j] + bias[r & 63];
  }
}

// ---------------- main: gather -> 7x7 interp -> WMMA GEMM ----------------
__global__ __launch_bounds__(128, 1) void sphereconv_main(
    const float* __restrict__ xt, const int* __restrict__ index,
    const float* __restrict__ itp_mat, const float* __restrict__ conv_w,
    float* __restrict__ outws)
{
  extern __shared__ char smem[];
  v2f*   wfrag = (v2f*)smem;                                      // [KSTEPS][4][32]
  float* itpS  = (float*)(smem + WFRAG_BYTES);                    // [WAVES][16][452]
  float* itpmS = (float*)(smem + WFRAG_BYTES + ITPS_BYTES);       // [WAVES][4][49]
  int*   idxS  = (int*)(smem + WFRAG_BYTES + ITPS_BYTES + ITPM_BYTES); // [WAVES][28]

  const int tid = threadIdx.x;

  // Stage weights into LDS in exact WMMA B-fragment order (shared by all waves):
  // W[(c*7+j)][o] == conv_w[o*448 + (c*7+j)].
  // Fragment (kk,n,lane): lanes 0-15 hold K=kbase,kbase+1 ; lanes 16-31 K=kbase+2,kbase+3.
  for (int s = tid; s < KSTEPS * 4 * 32; s += blockDim.x) {
    int lane = s & 31, n = (s >> 5) & 3, kk = s >> 7;
    int kbase = kk * 4 + ((lane & 16) ? 2 : 0);
    int o = n * 16 + (lane & 15);
    const float* wp = conv_w + o * KTOT + kbase;
    v2f w; w.x = wp[0]; w.y = wp[1];
    wfrag[s] = w;
  }
  __syncthreads();   // only cross-wave dependency in the whole kernel

  const int wave = tid >> 5, lane = tid & 31;
  const int hh = lane >> 4, mlane = lane & 15;
  float* itpW  = itpS  + wave * ITPS_FLOATS;
  float* itpmW = itpmS + wave * (4 * 49);
  int*   idxW  = idxS  + wave * 28;

  const int gw = blockIdx.x * WAVES + wave;
  const int tW = gridDim.x * WAVES;

  for (int grp = gw; grp < NGROUPS; grp += tW) {
    const int vbase = grp * 4;

    // ---- stage indices + per-vertex 7x7 matrices for this 4-vertex group ----
    if (lane < 28) {
      int vl = lane / 7, k = lane % 7;
      int v = vbase + vl; if (v >= V_) v = V_ - 1;
      int u = index[v * K_ + k];
      idxW[lane] = u;
      __builtin_prefetch(xt + (size_t)u * RR, 0, 1);        // global_prefetch_b8
      __builtin_prefetch(xt + (size_t)u * RR + 128, 0, 1);
    }
    for (int i = lane; i < 4 * 49; i += 32) {
      int vl = i / 49;
      int v = vbase + vl; if (v >= V_) v = V_ - 1;
      itpmW[i] = itp_mat[(size_t)v * 49 + (i % 49)];
    }
    wave_sync();

    // ---- interpolation: itpW[m=(vl*4+b)][c*7+j] = sum_k g * M[k][j] ----
    // Two M-rows per iteration; all 28 gather loads batched up front for MLP.
#pragma unroll 1
    for (int mm = 0; mm < 8; ++mm) {
      const int m0 = mm * 2;
      const int vl = m0 >> 2;
      const int b0 = m0 & 3, b1 = b0 + 1;
      const float* M = itpmW + vl * 49;
      const int*   U = idxW + vl * K_;

      float ga[2][7], gb[2][7];
#pragma unroll
      for (int p = 0; p < 2; ++p) {
        const int c = p * 32 + lane;
#pragma unroll
        for (int k = 0; k < K_; ++k) {
          const float* base = xt + (size_t)U[k] * RR + c;
          ga[p][k] = base[b0 * CIN];
          gb[p][k] = base[b1 * CIN];
        }
      }
#pragma unroll
      for (int p = 0; p < 2; ++p) {
        const int c = p * 32 + lane;
        float ta[7] = {0,0,0,0,0,0,0};
        float tb[7] = {0,0,0,0,0,0,0};
#pragma unroll
        for (int k = 0; k < K_; ++k) {
#pragma unroll
          for (int j = 0; j < 7; ++j) {
            float mkj = M[k * 7 + j];
            ta[j] += ga[p][k] * mkj;
            tb[j] += gb[p][k] * mkj;
          }
        }
        float* d0 = itpW + m0 * ITP_STRIDE + c * 7;
        float* d1 = d0 + ITP_STRIDE;
#pragma unroll
        for (int j = 0; j < 7; ++j) { d0[j] = ta[j]; d1[j] = tb[j]; }
      }
    }
    wave_sync();

    // ---- GEMM: D[16x64] = A[16x448] @ W[448x64], software-pipelined ----
    v8f acc[4];
    const v8f z = {0.f,0.f,0.f,0.f,0.f,0.f,0.f,0.f};
    acc[0]=z; acc[1]=z; acc[2]=z; acc[3]=z;

    const float* arow  = itpW + mlane * ITP_STRIDE + hh * 2;
    const v2f*   wbase = wfrag + lane;

    v2f a_cur = *(const v2f*)(arow);
    v2f b0 = wbase[0], b1 = wbase[32], b2 = wbase[64], b3 = wbase[96];

#pragma unroll 2
    for (int kk = 0; kk < KSTEPS - 1; ++kk) {
      // prefetch next k-step's fragments before issuing this step's WMMAs
      v2f a_nxt = *(const v2f*)(arow + (kk + 1) * 4);
      const v2f* wp = wbase + (kk + 1) * 128;
      v2f n0 = wp[0], n1 = wp[32], n2 = wp[64], n3 = wp[96];

      acc[0] = __builtin_amdgcn_wmma_f32_16x16x4_f32(false, a_cur, false, b0, (short)0, acc[0], false, false);
      acc[1] = __builtin_amdgcn_wmma_f32_16x16x4_f32(false, a_cur, false, b1, (short)0, acc[1], false, false);
      acc[2] = __builtin_amdgcn_wmma_f32_16x16x4_f32(false, a_cur, false, b2, (short)0, acc[2], false, false);
      acc[3] = __builtin_amdgcn_wmma_f32_16x16x4_f32(false, a_cur, false, b3, (short)0, acc[3], false, false);

      a_cur = a_nxt; b0 = n0; b1 = n1; b2 = n2; b3 = n3;
    }
    acc[0] = __builtin_amdgcn_wmma_f32_16x16x4_f32(false, a_cur, false, b0, (short)0, acc[0], false, false);
    acc[1] = __builtin_amdgcn_wmma_f32_16x16x4_f32(false, a_cur, false, b1, (short)0, acc[1], false, false);
    acc[2] = __builtin_amdgcn_wmma_f32_16x16x4_f32(false, a_cur, false, b2, (short)0, acc[2], false, false);
    acc[3] = __builtin_amdgcn_wmma_f32_16x16x4_f32(false, a_cur, false, b3, (short)0, acc[3], false, false);
    wave_sync();

    // ---- store D to outws(V,256); lanes 0-15 write 64B contiguous rows ----
    if (vbase + 3 < V_) {
      // fast path: no bounds checks (all groups except the very last one)
#pragma unroll
      for (int n = 0; n < 4; ++n) {
#pragma unroll
        for (int r = 0; r < 8; ++r) {
          const int mrow = r + 8 * hh;   // D layout: VGPR r -> M=r (lanes<16) / r+8
          const int vl = mrow >> 2, b = mrow & 3;
          outws[(size_t)(vbase + vl) * RR + b * CIN + n * 16 + mlane] = acc[n][r];
        }
      }
    } else {
#pragma unroll
      for (int n = 0; n < 4; ++n) {
#pragma unroll
        for (int r = 0; r < 8; ++r) {
          const int mrow = r + 8 * hh;
          const int vl = mrow >> 2, b = mrow & 3;
          const int v = vbase + vl;
          if (v < V_)
            outws[(size_t)v * RR + b * CIN + n * 16 + mlane] = acc[n][r];
        }
      }
    }
    wave_sync();   // before next iteration's staging overwrites idxW/itpmW/itpW
  }
}

extern "C" void kernel_launch(void* const* d_in, const int* in_sizes, int n_in,
                              void* d_out, int out_size, void* d_ws, size_t ws_size,
                              hipStream_t stream) {
  (void)in_sizes; (void)n_in; (void)out_size; (void)ws_size;
  const float* x      = (const float*)d_in[0];
  const int*   index  = (const int*)  d_in[1];
  const float* itpm   = (const float*)d_in[2];
  const float* conv_w = (const float*)d_in[3];
  const float* conv_b = (const float*)d_in[4];
  float* out   = (float*)d_out;
  float* xt    = (float*)d_ws;                   // V*256 floats
  float* outws = xt + (size_t)V_ * RR;           // V*256 floats

  dim3 tgrid((V_ + 31) / 32, RR / 32), tblk(32, 8);
  transpose_in_kernel<<<tgrid, tblk, 0, stream>>>(x, xt);

  hipFuncSetAttribute((const void*)sphereconv_main,
                      hipFuncAttributeMaxDynamicSharedMemorySize, SMEM_BYTES);
  sphereconv_main<<<512, 128, SMEM_BYTES, stream>>>(xt, index, itpm, conv_w, outws);

  transpose_out_kernel<<<tgrid, tblk, 0, stream>>>(outws, conv_b, out);
}